// TimeToFirstSpike_18399639896757
// MI455X (gfx1250) — compile-verified
//
#include <hip/hip_runtime.h>
#include <stdint.h>

#define TWIN 32

typedef float f4 __attribute__((ext_vector_type(4)));
typedef int v4i __attribute__((vector_size(4 * sizeof(int))));
typedef unsigned int u32;
typedef long long i64;

#define GAS __attribute__((address_space(1)))
#define LAS __attribute__((address_space(3)))

#if __has_builtin(__builtin_amdgcn_global_load_async_to_lds_b128)
#define HAS_ASYNC_LDS 1
#else
#define HAS_ASYNC_LDS 0
#endif

#if __has_builtin(__builtin_amdgcn_s_wait_asynccnt)
#define WAIT_ASYNC(N) do { __builtin_amdgcn_s_wait_asynccnt(N); asm volatile("" ::: "memory"); } while (0)
#else
#define WAIT_ASYNC(N) asm volatile("s_wait_asynccnt " #N ::: "memory")
#endif

__device__ __forceinline__ u32 umin2(u32 a, u32 b) { return a < b ? a : b; }
__device__ __forceinline__ u32 umax2(u32 a, u32 b) { return a > b ? a : b; }

// Monotone bijection float -> u32 so unsigned min/max == float min/max (total order).
__device__ __forceinline__ u32 fkey(float f) {
  u32 u = __float_as_uint(f);
  return u ^ ((u32)((int)u >> 31) | 0x80000000u);
}
__device__ __forceinline__ float keyf(u32 k) {
  u32 u = (k & 0x80000000u) ? (k ^ 0x80000000u) : ~k;
  return __uint_as_float(u);
}

__global__ void ttfs_init(u32* __restrict__ ws) {
  if (threadIdx.x == 0) { ws[0] = 0xFFFFFFFFu; ws[1] = 0u; }
}

// Pass 1: global min/max. Default (RT) loads on purpose: 19 MB input stays
// resident in the 192 MB L2 so the encode pass re-reads it from L2.
__global__ __launch_bounds__(256) void ttfs_minmax(const float* __restrict__ x, i64 n,
                                                   u32* __restrict__ ws) {
  __shared__ u32 bmin, bmax;
  if (threadIdx.x == 0) { bmin = 0xFFFFFFFFu; bmax = 0u; }
  __syncthreads();
  i64 tid = (i64)blockIdx.x * blockDim.x + threadIdx.x;
  i64 stride = (i64)gridDim.x * blockDim.x;
  u32 kmn = 0xFFFFFFFFu, kmx = 0u;
  i64 n4 = n >> 2;
  const f4* x4 = (const f4*)x;
  for (i64 i = tid; i < n4; i += stride) {
    f4 v = x4[i];
    u32 a = fkey(v.x), b = fkey(v.y), c = fkey(v.z), d = fkey(v.w);
    kmn = umin2(kmn, umin2(umin2(a, b), umin2(c, d)));
    kmx = umax2(kmx, umax2(umax2(a, b), umax2(c, d)));
  }
  for (i64 i = (n4 << 2) + tid; i < n; i += stride) {
    u32 a = fkey(x[i]);
    kmn = umin2(kmn, a);
    kmx = umax2(kmx, a);
  }
  atomicMin(&bmin, kmn);
  atomicMax(&bmax, kmx);
  __syncthreads();
  if (threadIdx.x == 0) { atomicMin(&ws[0], bmin); atomicMax(&ws[1], bmax); }
}

// One-hot emit for 4 elements: t_first = #{t : v < thr[t]} (thr strictly decreasing),
// then 32 NT b128 stores, one per time step (coalesced 512B/wave per step).
__device__ __forceinline__ void emit32v(f4 v, float mn, float scale, const float* thr,
                                        float* __restrict__ out, i64 n, i64 i4) {
  f4 vx = (v - mn) * scale + 0.001f;
  int c0 = 0, c1 = 0, c2 = 0, c3 = 0;
#pragma unroll
  for (int t = 0; t < TWIN; ++t) {
    float th = thr[t];
    c0 += vx.x < th;
    c1 += vx.y < th;
    c2 += vx.z < th;
    c3 += vx.w < th;
  }
  float* base = out + (i4 << 2);
#pragma unroll
  for (int t = 0; t < TWIN; ++t) {
    f4 o;
    o.x = (c0 == t) ? 1.0f : 0.0f;
    o.y = (c1 == t) ? 1.0f : 0.0f;
    o.z = (c2 == t) ? 1.0f : 0.0f;
    o.w = (c3 == t) ? 1.0f : 0.0f;
    __builtin_nontemporal_store(o, (f4*)(base + (i64)t * n));
  }
}

__global__ __launch_bounds__(256) void ttfs_encode(const float* __restrict__ x,
                                                   float* __restrict__ out, i64 n,
                                                   const u32* __restrict__ ws) {
  // Threshold table, exact reference formula; constant-folds at compile time.
  const float tau = (float)(32.0 / 6.90775527898213705205);  // float32(-T/ln(eps))
  float thr[TWIN];
#pragma unroll
  for (int t = 0; t < TWIN; ++t) thr[t] = expf(-(float)(t + 1) / tau);

  u32 kmn = ws[0], kmx = ws[1];
  float mn = keyf(kmn), mx = keyf(kmx);
  float scale = 0.999f / (mx - mn);  // (1-eps)/range

  i64 tid = (i64)blockIdx.x * blockDim.x + threadIdx.x;
  i64 nthreads = (i64)gridDim.x * blockDim.x;

  if ((n & 3) == 0) {
    i64 n4 = n >> 2;
    i64 ntiles = n4 >> 5;  // wave tiles: 32 lanes x float4 = 128 floats = 512B
    const f4* x4 = (const f4*)x;
#if HAS_ASYNC_LDS
    // Per-wave double-buffered async global->LDS pipeline (ASYNCcnt is per-wave;
    // each wave owns a disjoint 2x512B LDS slice, so no barriers needed).
    __shared__ f4 lbuf[2 * 8 * 32];  // 8 waves/block * 2 bufs * 512B = 8 KB
    int lane = threadIdx.x & 31;
    int wib = threadIdx.x >> 5;
    f4* buf[2] = {&lbuf[(wib * 2 + 0) * 32], &lbuf[(wib * 2 + 1) * 32]};
    i64 W = tid >> 5;
    i64 nwaves = nthreads >> 5;
    if (W < ntiles) {
      const f4* src = x4 + (W << 5) + lane;
      __builtin_amdgcn_global_load_async_to_lds_b128(
          (GAS v4i*)(v4i*)src, (LAS v4i*)(v4i*)(buf[0] + lane), 0, 0);
    }
    int cur = 0;
    for (i64 tile = W; tile < ntiles; tile += nwaves) {
      i64 nxt = tile + nwaves;
      if (nxt < ntiles) {
        const f4* src = x4 + (nxt << 5) + lane;
        __builtin_amdgcn_global_load_async_to_lds_b128(
            (GAS v4i*)(v4i*)src, (LAS v4i*)(v4i*)(buf[cur ^ 1] + lane), 0, 0);
        WAIT_ASYNC(1);
      } else {
        WAIT_ASYNC(0);
      }
      f4 v = buf[cur][lane];  // ds_load_b128
      emit32v(v, mn, scale, thr, out, n, (tile << 5) + lane);
      cur ^= 1;
    }
#else
    i64 nv = ntiles << 5;
    for (i64 i4 = tid; i4 < nv; i4 += nthreads) {
      f4 v = x4[i4];  // RT load: hits L2 (primed by minmax pass)
      emit32v(v, mn, scale, thr, out, n, i4);
    }
#endif
  }

  // Scalar remainder (and full fallback when n % 4 != 0).
  i64 start = ((n & 3) == 0) ? (((n >> 2) >> 5) << 7) : 0;
  for (i64 i = start + tid; i < n; i += nthreads) {
    float vx = (x[i] - mn) * scale + 0.001f;
    int c = 0;
#pragma unroll
    for (int t = 0; t < TWIN; ++t) c += vx < thr[t];
#pragma unroll
    for (int t = 0; t < TWIN; ++t)
      __builtin_nontemporal_store((c == t) ? 1.0f : 0.0f, out + (i64)t * n + i);
  }
}

extern "C" void kernel_launch(void* const* d_in, const int* in_sizes, int n_in, void* d_out,
                              int out_size, void* d_ws, size_t ws_size, hipStream_t stream) {
  (void)n_in; (void)out_size; (void)ws_size;
  const float* x = (const float*)d_in[0];
  float* out = (float*)d_out;
  i64 n = (i64)in_sizes[0];
  u32* ws = (u32*)d_ws;

  ttfs_init<<<1, 64, 0, stream>>>(ws);
  ttfs_minmax<<<512, 256, 0, stream>>>(x, n, ws);

  // ~4 wave-tiles per wave so the async double-buffer actually pipelines.
  i64 ntiles = (n >> 2) >> 5;
  i64 waves = (ntiles + 3) / 4;
  i64 blocks = (waves + 7) / 8;  // 8 waves per 256-thread block
  if (blocks < 1) blocks = 1;
  if (blocks > 65535) blocks = 65535;
  ttfs_encode<<<(int)blocks, 256, 0, stream>>>(x, out, n, ws);
}